// CondTransportKernel_4131758539282
// MI455X (gfx1250) — compile-verified
//
#include <hip/hip_runtime.h>
#include <hip/hip_bf16.h>

// ---------------------------------------------------------------------------
// CondTransport loss on MI455X (gfx1250).
//
//  * Pairwise-distance dot products (rank-4 GEMMs) -> V_WMMA_F32_16X16X4_F32.
//  * exp() on the VALU trans pipe (co-executes with WMMA).
//  * K^{-1} Z via fixed-iteration matrix-free CG (K is SPD: RBF + nugget);
//    K tiles are recomputed per matvec (268 MB matrix > 192 MB L2, recompute
//    is cheaper than streaming from HBM every iteration).
//
// Workspace usage: ~24*N floats (~790 KB) — assumed <= ws_size.
// ---------------------------------------------------------------------------

#define NPTS       8192
#define NT         (NPTS / 16)     // 512 16x16 tiles per dimension
#define NUGGET     1e-4f
#define REG_LAMBDA 1e-5f
#define CG_ITERS   64
#define JCHUNKS    8               // J-strips per tile-row in the matvec

typedef float v2f __attribute__((ext_vector_type(2)));
typedef float v8f __attribute__((ext_vector_type(8)));

// ---- WMMA: C(16x16,f32) = A(16x4,f32) * B(4x16,f32) -----------------------
// Lane l holds: A row (l&15), K components {2*(l>>4), 2*(l>>4)+1} (2 VGPRs);
// same layout for B columns.  C: VGPR v, lane l -> M = v + 8*(l>>4), N = l&15.
__device__ __forceinline__ v8f wmma_dot4(v2f a, v2f b) {
  v8f c = {};
  return __builtin_amdgcn_wmma_f32_16x16x4_f32(
      /*neg_a=*/false, a, /*neg_b=*/false, b,
      /*c_mod=*/(short)0, c, /*reuse_a=*/false, /*reuse_b=*/false);
}

__device__ __forceinline__ float wave_sum32(float v) {
#pragma unroll
  for (int m = 16; m >= 1; m >>= 1) v += __shfl_xor(v, m, 32);
  return v;
}
// Reduction within each 16-lane half (xor masks < 16 keep lane>>4 fixed).
__device__ __forceinline__ float half_sum16(float v) {
#pragma unroll
  for (int m = 8; m >= 1; m >>= 1) v += __shfl_xor(v, m, 32);
  return v;
}

// ---- pack 4-D points + half-norms -----------------------------------------
__global__ __launch_bounds__(256) void precompute_kernel(
    const float2* __restrict__ Xmu, const float2* __restrict__ Yeta,
    const float2* __restrict__ Ymu, const float2* __restrict__ Zp,
    float* __restrict__ Pmap, float* __restrict__ PY, float* __restrict__ PX,
    float* __restrict__ HNmap, float* __restrict__ HNY, float* __restrict__ HNX) {
  int i = blockIdx.x * blockDim.x + threadIdx.x;
  if (i >= NPTS) return;
  float2 x = Xmu[i], ye = Yeta[i], ym = Ymu[i], z = Zp[i];
  float4 pm = make_float4(x.x, x.y, ye.x + z.x, ye.y + z.y);  // map_vec
  float4 py = make_float4(x.x, x.y, ym.x, ym.y);              // target Y
  float4 px = make_float4(x.x, x.y, ye.x, ye.y);              // fit X
  ((float4*)Pmap)[i] = pm;
  ((float4*)PY)[i]   = py;
  ((float4*)PX)[i]   = px;
  HNmap[i] = 0.5f * (pm.x*pm.x + pm.y*pm.y + pm.z*pm.z + pm.w*pm.w);
  HNY[i]   = 0.5f * (py.x*py.x + py.y*py.y + py.z*py.z + py.w*py.w);
  HNX[i]   = 0.5f * (px.x*px.x + px.y*px.y + px.z*px.z + px.w*px.w);
}

// ---- MMD: sum over 3 Gram matrices of exp(dot - hna - hnb) ----------------
// acc[0]=S_ZZ, acc[1]=S_ZY, acc[2]=S_YY
__global__ __launch_bounds__(256) void mmd_kernel(
    const float* __restrict__ Pmap, const float* __restrict__ PY,
    const float* __restrict__ HNmap, const float* __restrict__ HNY,
    float* __restrict__ acc) {
  const int lane   = threadIdx.x & 31;
  const int wave   = blockIdx.x * (blockDim.x >> 5) + (threadIdx.x >> 5);
  const int nwaves = gridDim.x * (blockDim.x >> 5);
  const int m      = lane & 15;
  const int half   = lane >> 4;
  float s0 = 0.f, s1 = 0.f, s2 = 0.f;

  for (int t = wave; t < 3 * NT * NT; t += nwaves) {   // wave-uniform loop
    const int g = t / (NT * NT);
    const int r = t - g * (NT * NT);
    const int I = r / NT, J = r - I * NT;
    const float* PA = (g == 2) ? PY : Pmap;
    const float* PB = (g == 0) ? Pmap : PY;
    const float* HA = (g == 2) ? HNY : HNmap;
    const float* HB = (g == 0) ? HNmap : HNY;

    v2f a = *(const v2f*)(PA + (size_t)(I * 16 + m) * 4 + half * 2);
    v2f b = *(const v2f*)(PB + (size_t)(J * 16 + m) * 4 + half * 2);
    v8f c = wmma_dot4(a, b);

    const float hb    = HB[J * 16 + m];
    const int   mbase = I * 16 + 8 * half;
    float local = 0.f;
#pragma unroll
    for (int v = 0; v < 8; ++v)
      local += __expf(c[v] - HA[mbase + v] - hb);

    if (g == 0) s0 += local; else if (g == 1) s1 += local; else s2 += local;
  }
  s0 = wave_sum32(s0); s1 = wave_sum32(s1); s2 = wave_sum32(s2);
  if (lane == 0) {
    atomicAdd(&acc[0], s0);
    atomicAdd(&acc[1], s1);
    atomicAdd(&acc[2], s2);
  }
}

// ---- CG matvec: q += K_fit(tile-recomputed) @ p  (nugget added later) -----
// grid = (NT/8, JCHUNKS), block = 256 (8 waves; each wave owns one tile-row I
// and one strip of J tiles; A tile + row norms stay resident across strip).
__global__ __launch_bounds__(256) void matvec_kernel(
    const float* __restrict__ PX, const float* __restrict__ HNX,
    const float* __restrict__ p, float* __restrict__ q) {
  const int lane = threadIdx.x & 31;
  const int m    = lane & 15;
  const int half = lane >> 4;
  const int I    = blockIdx.x * 8 + (threadIdx.x >> 5);
  const int J0   = blockIdx.y * (NT / JCHUNKS);
  const int J1   = J0 + (NT / JCHUNKS);

  const v2f a     = *(const v2f*)(PX + (size_t)(I * 16 + m) * 4 + half * 2);
  const int mbase = I * 16 + 8 * half;
  float ha[8];
#pragma unroll
  for (int v = 0; v < 8; ++v) ha[v] = HNX[mbase + v];

  float acc0[8] = {0,0,0,0,0,0,0,0};
  float acc1[8] = {0,0,0,0,0,0,0,0};

  for (int J = J0; J < J1; ++J) {                      // wave-uniform loop
    v2f b = *(const v2f*)(PX + (size_t)(J * 16 + m) * 4 + half * 2);
    v8f c = wmma_dot4(a, b);
    const int   col = J * 16 + m;
    const float hb  = HNX[col];
    const float2 pc = ((const float2*)p)[col];
#pragma unroll
    for (int v = 0; v < 8; ++v) {
      float e = __expf(c[v] - ha[v] - hb);
      acc0[v] += e * pc.x;
      acc1[v] += e * pc.y;
    }
  }
#pragma unroll
  for (int v = 0; v < 8; ++v) {
    float r0 = half_sum16(acc0[v]);
    float r1 = half_sum16(acc1[v]);
    if (m == 0) {                                      // lane 0 and lane 16
      atomicAdd(&q[(size_t)(mbase + v) * 2 + 0], r0);
      atomicAdd(&q[(size_t)(mbase + v) * 2 + 1], r1);
    }
  }
}

// ---- q += nugget*p ; pAp[c] = sum p.q --------------------------------------
__global__ __launch_bounds__(256) void finish_dot_kernel(
    const float* __restrict__ p, float* __restrict__ q, float* __restrict__ pAp) {
  int i = blockIdx.x * blockDim.x + threadIdx.x;
  float d0 = 0.f, d1 = 0.f;
  if (i < NPTS) {
    float2 pv = ((const float2*)p)[i];
    float2 qv = ((const float2*)q)[i];
    qv.x += NUGGET * pv.x;
    qv.y += NUGGET * pv.y;
    ((float2*)q)[i] = qv;
    d0 = pv.x * qv.x;
    d1 = pv.y * qv.y;
  }
  d0 = wave_sum32(d0); d1 = wave_sum32(d1);
  if ((threadIdx.x & 31) == 0) {
    atomicAdd(&pAp[0], d0);
    atomicAdd(&pAp[1], d1);
  }
}

__global__ void cg_alpha_kernel(const float* rs, const float* pAp, float* alpha) {
  if (threadIdx.x == 0) {
#pragma unroll
    for (int c = 0; c < 2; ++c) {
      float d = pAp[c];
      alpha[c] = (d != 0.f) ? rs[c] / d : 0.f;
    }
  }
}

// x += alpha p ; r -= alpha q ; rsnew[c] = ||r_c||^2
__global__ __launch_bounds__(256) void cg_update_xr_kernel(
    const float* __restrict__ alpha, const float* __restrict__ p,
    const float* __restrict__ q, float* __restrict__ x, float* __restrict__ r,
    float* __restrict__ rsnew) {
  int i = blockIdx.x * blockDim.x + threadIdx.x;
  float d0 = 0.f, d1 = 0.f;
  if (i < NPTS) {
    float a0 = alpha[0], a1 = alpha[1];
    float2 pv = ((const float2*)p)[i];
    float2 qv = ((const float2*)q)[i];
    float2 xv = ((const float2*)x)[i];
    float2 rv = ((const float2*)r)[i];
    xv.x += a0 * pv.x;  xv.y += a1 * pv.y;
    rv.x -= a0 * qv.x;  rv.y -= a1 * qv.y;
    ((float2*)x)[i] = xv;
    ((float2*)r)[i] = rv;
    d0 = rv.x * rv.x;   d1 = rv.y * rv.y;
  }
  d0 = wave_sum32(d0); d1 = wave_sum32(d1);
  if ((threadIdx.x & 31) == 0) {
    atomicAdd(&rsnew[0], d0);
    atomicAdd(&rsnew[1], d1);
  }
}

__global__ void cg_beta_kernel(float* rs, const float* rsnew, float* beta) {
  if (threadIdx.x == 0) {
#pragma unroll
    for (int c = 0; c < 2; ++c) {
      float o = rs[c];
      beta[c] = (o != 0.f) ? rsnew[c] / o : 0.f;
      rs[c] = rsnew[c];
    }
  }
}

__global__ __launch_bounds__(256) void cg_update_p_kernel(
    const float* __restrict__ beta, const float* __restrict__ r,
    float* __restrict__ p) {
  int i = blockIdx.x * blockDim.x + threadIdx.x;
  if (i >= NPTS) return;
  float2 rv = ((const float2*)r)[i];
  float2 pv = ((const float2*)p)[i];
  pv.x = rv.x + beta[0] * pv.x;
  pv.y = rv.y + beta[1] * pv.y;
  ((float2*)p)[i] = pv;
}

// x = 0; r = p = Z; rs[c] = ||Z_c||^2  (rs pre-zeroed)
__global__ __launch_bounds__(256) void cg_init_kernel(
    const float* __restrict__ Z, float* __restrict__ x, float* __restrict__ r,
    float* __restrict__ p, float* __restrict__ rs) {
  int i = blockIdx.x * blockDim.x + threadIdx.x;
  float d0 = 0.f, d1 = 0.f;
  if (i < NPTS) {
    float2 zv = ((const float2*)Z)[i];
    ((float2*)x)[i] = make_float2(0.f, 0.f);
    ((float2*)r)[i] = zv;
    ((float2*)p)[i] = zv;
    d0 = zv.x * zv.x;  d1 = zv.y * zv.y;
  }
  d0 = wave_sum32(d0); d1 = wave_sum32(d1);
  if ((threadIdx.x & 31) == 0) {
    atomicAdd(&rs[0], d0);
    atomicAdd(&rs[1], d1);
  }
}

// regacc += sum(Z .* w)
__global__ __launch_bounds__(256) void final_dot_kernel(
    const float* __restrict__ Z, const float* __restrict__ x,
    float* __restrict__ regacc) {
  int i = blockIdx.x * blockDim.x + threadIdx.x;
  float d = 0.f;
  if (i < NPTS) {
    float2 zv = ((const float2*)Z)[i];
    float2 xv = ((const float2*)x)[i];
    d = zv.x * xv.x + zv.y * xv.y;
  }
  d = wave_sum32(d);
  if ((threadIdx.x & 31) == 0) atomicAdd(regacc, d);
}

__global__ void finalize_kernel(const float* __restrict__ acc,
                                const float* __restrict__ regacc,
                                float* __restrict__ out) {
  if (threadIdx.x == 0) {
    const float invN2 = 1.0f / ((float)NPTS * (float)NPTS);
    float mmd = (acc[0] - 2.0f * acc[1] + acc[2]) * invN2;
    out[0] = mmd + REG_LAMBDA * regacc[0];
  }
}

extern "C" void kernel_launch(void* const* d_in, const int* in_sizes, int n_in,
                              void* d_out, int out_size, void* d_ws, size_t ws_size,
                              hipStream_t stream) {
  (void)in_sizes; (void)n_in; (void)out_size; (void)ws_size;
  const float2* Xmu  = (const float2*)d_in[0];
  const float2* Yeta = (const float2*)d_in[1];
  const float2* Ymu  = (const float2*)d_in[2];
  const float2* Zp   = (const float2*)d_in[3];
  const float*  Zf   = (const float*)d_in[3];

  // -------- workspace layout (floats) --------
  float* ws    = (float*)d_ws;
  float* Pmap  = ws;                  // N*4
  float* PY    = Pmap + NPTS * 4;     // N*4
  float* PX    = PY   + NPTS * 4;     // N*4
  float* HNmap = PX   + NPTS * 4;     // N
  float* HNY   = HNmap + NPTS;        // N
  float* HNX   = HNY   + NPTS;        // N
  float* xw    = HNX   + NPTS;        // N*2  (solution w)
  float* rw    = xw    + NPTS * 2;    // N*2
  float* pw    = rw    + NPTS * 2;    // N*2
  float* qw    = pw    + NPTS * 2;    // N*2
  float* sc    = qw    + NPTS * 2;    // scalars
  float* mmdAcc = sc + 0;             // [3]
  float* rs     = sc + 3;             // [2]
  float* pAp    = sc + 5;             // [2]
  float* rsnew  = sc + 7;             // [2]
  float* alpha  = sc + 9;             // [2]
  float* beta   = sc + 11;            // [2]
  float* regacc = sc + 13;            // [1]

  const int ethreads = 256;
  const int eblocks  = NPTS / ethreads;        // 32

  hipMemsetAsync(sc, 0, 14 * sizeof(float), stream);

  precompute_kernel<<<eblocks, ethreads, 0, stream>>>(
      Xmu, Yeta, Ymu, Zp, Pmap, PY, PX, HNmap, HNY, HNX);

  // -------- MMD (3 Gram matrices, WMMA tile dot + exp) --------
  mmd_kernel<<<2048, 256, 0, stream>>>(Pmap, PY, HNmap, HNY, mmdAcc);

  // -------- CG: K_fit w = Z --------
  cg_init_kernel<<<eblocks, ethreads, 0, stream>>>(Zf, xw, rw, pw, rs);

  dim3 mvGrid(NT / 8, JCHUNKS);
  for (int it = 0; it < CG_ITERS; ++it) {
    hipMemsetAsync(qw, 0, NPTS * 2 * sizeof(float), stream);
    hipMemsetAsync(pAp, 0, 4 * sizeof(float), stream);   // pAp + rsnew
    matvec_kernel<<<mvGrid, 256, 0, stream>>>(PX, HNX, pw, qw);
    finish_dot_kernel<<<eblocks, ethreads, 0, stream>>>(pw, qw, pAp);
    cg_alpha_kernel<<<1, 32, 0, stream>>>(rs, pAp, alpha);
    cg_update_xr_kernel<<<eblocks, ethreads, 0, stream>>>(alpha, pw, qw, xw, rw, rsnew);
    cg_beta_kernel<<<1, 32, 0, stream>>>(rs, rsnew, beta);
    cg_update_p_kernel<<<eblocks, ethreads, 0, stream>>>(beta, rw, pw);
  }

  // -------- combine --------
  final_dot_kernel<<<eblocks, ethreads, 0, stream>>>(Zf, xw, regacc);
  finalize_kernel<<<1, 32, 0, stream>>>(mmdAcc, regacc, (float*)d_out);
}